// LocalSpatialEncoding_68745246540410
// MI455X (gfx1250) — compile-verified
//
#include <hip/hip_runtime.h>
#include <hip/hip_bf16.h>

// ---------------------------------------------------------------------------
// LocalSpatialEncoding for MI455X (gfx1250, wave32)
//   B=4, N=32768, K=16, D=8 ; out (B, 2D, N, K) f32
//   Pass 1: per-position x = W*feat + b, accumulate f64 sum/sumsq per channel
//   Pass 2: finalize BN scale/shift
//   Pass 3: V_WMMA_F32_16X16X4_F32 tile per (b,n): 16 k-positions x 10 chans
// ---------------------------------------------------------------------------

constexpr int kB = 4;
constexpr int kN = 32768;   // 2^15
constexpr int kK = 16;
constexpr int kD = 8;
constexpr long long kP = (long long)kB * kN * kK;  // 2,097,152 positions
constexpr int S_BLOCKS  = 512;
constexpr int S_THREADS = 256;
constexpr float kEPS = 1e-6f;

typedef __attribute__((ext_vector_type(2))) float v2f;
typedef __attribute__((ext_vector_type(4))) float v4f;
typedef __attribute__((ext_vector_type(8))) float v8f;

// ---------------- Pass 1: batch-norm statistics (f64 accumulation) ----------
__global__ void lse_stats_kernel(const float* __restrict__ coords,
                                 const long long* __restrict__ knn_idx,
                                 const float* __restrict__ knn_dist,
                                 const int* __restrict__ mask,
                                 const float* __restrict__ conv_w,
                                 const float* __restrict__ conv_b,
                                 double* __restrict__ partials) {
    // weights live in registers for the whole loop (statically indexed)
    float w[80], bias[kD];
#pragma unroll
    for (int i = 0; i < 80; ++i) w[i] = conv_w[i];
#pragma unroll
    for (int i = 0; i < kD; ++i) bias[i] = conv_b[i];

    double s[kD], ss[kD];
#pragma unroll
    for (int d = 0; d < kD; ++d) { s[d] = 0.0; ss[d] = 0.0; }

    const int stride = gridDim.x * blockDim.x;
    for (long long p = blockIdx.x * blockDim.x + threadIdx.x; p < kP; p += stride) {
        const int kq = (int)(p & (kK - 1));
        const int bn = (int)(p >> 4);           // b*N + n
        const int b  = bn >> 15;                // N = 2^15
        const float* c = coords + (size_t)bn * 3;
        const float cx = c[0], cy = c[1], cz = c[2];
        long long j = knn_idx[p];
        if (j < 0) j = 0;
        const float* q = coords + ((size_t)b * kN + (size_t)j) * 3;
        const float qx = q[0], qy = q[1], qz = q[2];
        const float dist = (mask[bn] == 0) ? __builtin_inff()
                                           : knn_dist[(size_t)bn * kK + kq];
        const float f[10] = {cx, cy, cz, qx, qy, qz, cx - qx, cy - qy, cz - qz, dist};
#pragma unroll
        for (int d = 0; d < kD; ++d) {
            float x = bias[d];
#pragma unroll
            for (int cc = 0; cc < 10; ++cc) x = fmaf(w[d * 10 + cc], f[cc], x);
            s[d]  += (double)x;
            ss[d] += (double)x * (double)x;
        }
    }

    // block tree-reduction of the 16 doubles
    __shared__ double red[S_THREADS];
#pragma unroll
    for (int i = 0; i < 2 * kD; ++i) {
        const double v = (i < kD) ? s[i] : ss[i - kD];
        __syncthreads();
        red[threadIdx.x] = v;
        __syncthreads();
        for (int off = S_THREADS / 2; off > 0; off >>= 1) {
            if (threadIdx.x < off) red[threadIdx.x] += red[threadIdx.x + off];
            __syncthreads();
        }
        if (threadIdx.x == 0) partials[(size_t)blockIdx.x * (2 * kD) + i] = red[0];
    }
}

// ---------------- Pass 2: fold partials -> scale/shift ----------------------
__global__ void lse_finalize_kernel(const double* __restrict__ partials,
                                    const float* __restrict__ gamma,
                                    const float* __restrict__ beta,
                                    float* __restrict__ scale_shift) {
    const int t = threadIdx.x;  // 16 threads
    double acc = 0.0;
    for (int blk = 0; blk < S_BLOCKS; ++blk) acc += partials[(size_t)blk * 16 + t];
    __shared__ double tot[16];
    tot[t] = acc;
    __syncthreads();
    if (t < kD) {
        const double cnt  = (double)kP;
        const double mean = tot[t] / cnt;
        const double var  = tot[t + kD] / cnt - mean * mean;
        const double inv  = 1.0 / sqrt(var + (double)kEPS);
        const float  sc   = gamma[t] * (float)inv;
        const float  sh   = beta[t] - (float)mean * sc;
        scale_shift[t]      = sc;
        scale_shift[kD + t] = sh;
    }
}

// ---------------- Pass 3: WMMA projection + BN + ReLU + feature concat ------
__global__ void __launch_bounds__(256)
lse_encode_kernel(const float* __restrict__ coords,
                  const float* __restrict__ features,
                  const long long* __restrict__ knn_idx,
                  const float* __restrict__ knn_dist,
                  const int* __restrict__ mask,
                  const float* __restrict__ conv_w,
                  const float* __restrict__ conv_b,
                  const float* __restrict__ scale_shift,
                  float* __restrict__ out) {
    const int tid  = blockIdx.x * blockDim.x + threadIdx.x;
    const int wid  = tid >> 5;             // wave id == b*N + n
    const int lane = threadIdx.x & 31;
    const int b    = wid >> 15;            // N = 2^15
    const int n    = wid & (kN - 1);
    const int bn   = wid;

    const int kq = lane & 15;              // A-matrix row M = k ; B column N = d
    const int h  = lane >> 4;              // which half-wave (selects K pair)

    // --- feat components for position (b, n, kq) -----------------------------
    const float* c = coords + (size_t)bn * 3;
    const float cx = c[0], cy = c[1], cz = c[2];
    long long j = knn_idx[(size_t)bn * kK + kq];
    if (j < 0) j = 0;
    const float* q = coords + ((size_t)b * kN + (size_t)j) * 3;
    const float qx = q[0], qy = q[1], qz = q[2];
    const float dist = (mask[bn] == 0) ? __builtin_inff()
                                       : knn_dist[(size_t)bn * kK + kq];
    const float dx = cx - qx, dy = cy - qy, dz = cz - qz;

    // --- A operands: lane holds M=kq ; VGPR0/1 = channels (2h, 2h+1) of each
    //     K-group {0..3},{4..7},{8..11}; channels 10,11 are zero pad ----------
    v2f a0, a1, a2;
    a0.x = h ? cz  : cx;   a0.y = h ? qx  : cy;
    a1.x = h ? dx  : qy;   a1.y = h ? dy  : qz;
    a2.x = h ? 0.f : dz;   a2.y = h ? 0.f : dist;

    // --- B operands: lane holds column d=kq (d<8 valid); direct addressed ----
    const int  d  = kq;
    const bool dv = (d < kD);
    const float* wp = conv_w + d * 10 + 2 * h;
    v2f b0, b1, b2;
    b0.x = dv ? wp[0] : 0.f;          b0.y = dv ? wp[1] : 0.f;
    b1.x = dv ? wp[4] : 0.f;          b1.y = dv ? wp[5] : 0.f;
    b2.x = (dv && !h) ? wp[8] : 0.f;  b2.y = (dv && !h) ? wp[9] : 0.f;

    // --- C seeded with conv bias (bias depends only on column d) -------------
    const float bias_d = dv ? conv_b[d] : 0.f;
    v8f acc;
#pragma unroll
    for (int v = 0; v < 8; ++v) acc[v] = bias_d;

    // --- 3x V_WMMA_F32_16X16X4_F32, K-groups of 4 channels -------------------
    acc = __builtin_amdgcn_wmma_f32_16x16x4_f32(false, a0, false, b0,
                                                (short)0, acc, false, false);
    acc = __builtin_amdgcn_wmma_f32_16x16x4_f32(false, a1, false, b1,
                                                (short)0, acc, false, false);
    acc = __builtin_amdgcn_wmma_f32_16x16x4_f32(false, a2, false, b2,
                                                (short)0, acc, false, false);

    // --- BN affine + ReLU, store: lane owns (b, d, n, k = h*8 + v) -----------
    if (dv) {
        const float sc = scale_shift[d];
        const float sh = scale_shift[kD + d];
        v4f o0, o1;
#pragma unroll
        for (int v = 0; v < 4; ++v) {
            o0[v] = fmaxf(0.f, fmaf(acc[v],     sc, sh));
            o1[v] = fmaxf(0.f, fmaf(acc[v + 4], sc, sh));
        }
        const size_t base = ((((size_t)b * 16 + d) * kN + n) * kK) + (size_t)h * 8;
        *(v4f*)(out + base)     = o0;
        *(v4f*)(out + base + 4) = o1;
    }

    // --- feature-broadcast half: out[b, 8+dF, n, :] = features[b, dF, n, 0] --
    const int dF = lane >> 2;      // 0..7
    const int kF = lane & 3;       // quad of k
    const float fv = features[((size_t)b * kD + dF) * kN + n];
    const size_t fbase = ((((size_t)b * 16 + 8 + dF) * kN + n) * kK) + (size_t)kF * 4;
    v4f fo = {fv, fv, fv, fv};
    *(v4f*)(out + fbase) = fo;
}

// ---------------------------------------------------------------------------
extern "C" void kernel_launch(void* const* d_in, const int* in_sizes, int n_in,
                              void* d_out, int out_size, void* d_ws, size_t ws_size,
                              hipStream_t stream) {
    const float*     coords   = (const float*)d_in[0];
    const float*     features = (const float*)d_in[1];
    const long long* knn_idx  = (const long long*)d_in[2];  // int64
    const float*     knn_dist = (const float*)d_in[3];
    const int*       mask     = (const int*)d_in[4];
    const float*     conv_w   = (const float*)d_in[5];
    const float*     conv_b   = (const float*)d_in[6];
    const float*     bn_gamma = (const float*)d_in[7];
    const float*     bn_beta  = (const float*)d_in[8];
    float*           out      = (float*)d_out;

    double* partials    = (double*)d_ws;                                  // 512*16 f64
    float*  scale_shift = (float*)((char*)d_ws + (size_t)S_BLOCKS * 16 * sizeof(double));

    lse_stats_kernel<<<S_BLOCKS, S_THREADS, 0, stream>>>(
        coords, knn_idx, knn_dist, mask, conv_w, conv_b, partials);

    lse_finalize_kernel<<<1, 16, 0, stream>>>(partials, bn_gamma, bn_beta, scale_shift);

    const int total_waves = kB * kN;          // one wave per (b, n)
    lse_encode_kernel<<<total_waves / 8, 256, 0, stream>>>(
        coords, features, knn_idx, knn_dist, mask, conv_w, conv_b, scale_shift, out);
}